// DecGreenNet_product_CP4_90967407329574
// MI455X (gfx1250) — compile-verified
//
#include <hip/hip_runtime.h>
#include <math.h>

typedef __attribute__((ext_vector_type(2))) float v2f;
typedef __attribute__((ext_vector_type(8))) float v8f;

#define HID 1024
#define P4 4096

// Branch-free tanh: hardware v_tanh_f32 if available, else exp2/rcp identity.
__device__ __forceinline__ float fast_tanhf(float x) {
#if __has_builtin(__builtin_amdgcn_tanhf)
  return __builtin_amdgcn_tanhf(x);
#else
  // tanh(x) = sign(x) * (1 - 2/(1 + exp2(2*log2e*|x|)))
  float ax = __builtin_fabsf(x);
  float e = __builtin_amdgcn_exp2f(ax * 2.8853900817779268f);
  float t = 1.0f - 2.0f * __builtin_amdgcn_rcpf(e + 1.0f);
  return __builtin_copysignf(t, x);
#endif
}

// ---------------------------------------------------------------------------
// Phase 1: per-factor S_i[8,8] = sum_q y_q * (tanh(qx_q*Wa + ba) @ Wb + bb)
// ---------------------------------------------------------------------------
__global__ __launch_bounds__(128) void quad_S_kernel(
    const float* __restrict__ eq, const float* __restrict__ qx,
    const float* __restrict__ Wa, const float* __restrict__ ba,
    const float* __restrict__ Wb, const float* __restrict__ bb,
    float* __restrict__ Sout) {
  __shared__ float t[64][128];
  __shared__ float y[64];
  const int h = threadIdx.x;  // 0..127
  const float wa = Wa[h], bah = ba[h];
  for (int q = 0; q < 64; ++q) t[q][h] = fast_tanhf(qx[q] * wa + bah);
  if (h < 64) y[h] = eq[0] * sinf(3.14159265358979323846f * qx[h]);
  __syncthreads();
  if (h < 64) {
    const float bbo = bb[h];
    float s = 0.0f;
    for (int q = 0; q < 64; ++q) {
      float inner = bbo;
      for (int hh = 0; hh < 128; ++hh) inner += t[q][hh] * Wb[hh * 64 + h];
      s += y[q] * inner;
    }
    Sout[h] = s;  // S[p,r] flat, p*8+r
  }
}

// ---------------------------------------------------------------------------
// Phase 2: rhs[b,d,f,m] = sum_x S0[b,x] S1[d,x] S2[f,x] S3[m,x]
// ---------------------------------------------------------------------------
__global__ __launch_bounds__(256) void rhs_kernel(const float* __restrict__ S,
                                                  float* __restrict__ rhs) {
  __shared__ float s0[64], s1[64], s2[64], s3[64];
  const int t = threadIdx.x;
  if (t < 64) {
    s0[t] = S[t]; s1[t] = S[64 + t]; s2[t] = S[128 + t]; s3[t] = S[192 + t];
  }
  __syncthreads();
  for (int o = t; o < P4; o += 256) {
    const int b = o >> 9, d = (o >> 6) & 7, f = (o >> 3) & 7, m = o & 7;
    float s = 0.0f;
#pragma unroll
    for (int x = 0; x < 8; ++x)
      s += s0[b * 8 + x] * s1[d * 8 + x] * s2[f * 8 + x] * s3[m * 8 + x];
    rhs[o] = s;
  }
}

// ---------------------------------------------------------------------------
// Phase 3: v[k] = W2[k,:]·rhs  (16MB stream -> b128 loads), block HID -> c=b2·rhs
// ---------------------------------------------------------------------------
__global__ __launch_bounds__(256) void w2_rhs_kernel(
    const float* __restrict__ W2, const float* __restrict__ b2,
    const float* __restrict__ rhs, float* __restrict__ v, float* __restrict__ c) {
  __shared__ float red[256];
  const int t = threadIdx.x;
  const float* row = (blockIdx.x < HID) ? (W2 + (size_t)blockIdx.x * P4) : b2;
  const float4* row4 = (const float4*)row;
  const float4* rhs4 = (const float4*)rhs;
  float s = 0.0f;
#pragma unroll
  for (int i = t; i < P4 / 4; i += 256) {
    float4 a = row4[i];
    float4 b = rhs4[i];
    s += a.x * b.x + a.y * b.y + a.z * b.z + a.w * b.w;
  }
  red[t] = s;
  __syncthreads();
  for (int off = 128; off > 0; off >>= 1) {
    if (t < off) red[t] += red[t + off];
    __syncthreads();
  }
  if (t == 0) {
    if (blockIdx.x < HID) v[blockIdx.x] = red[0];
    else *c = red[0];
  }
}

// ---------------------------------------------------------------------------
// Phase 4: out[n] = sum_k tanh(input[n]·W1[:,k] + b1[k]) * v[k] + c
// One wave per 16-row tile; V_WMMA_F32_16X16X4_F32 (K=3 padded to 4).
// W1/b1/v staged in LDS; loop body is branch-free (cndmask, no exec churn).
// C/D layout: VGPR r, lane l -> row = r + 8*(l>=16), col = l%16.
// ---------------------------------------------------------------------------
__global__ __launch_bounds__(256) void fused_out_kernel(
    const float* __restrict__ input, const float* __restrict__ W1,
    const float* __restrict__ b1, const float* __restrict__ v,
    const float* __restrict__ cptr, float* __restrict__ out, int n) {
  __shared__ float sW1[3 * HID];
  __shared__ float sB1[HID];
  __shared__ float sV[HID];

  const int tid = threadIdx.x;
  for (int i = tid; i < 3 * HID; i += 256) sW1[i] = W1[i];
  for (int i = tid; i < HID; i += 256) { sB1[i] = b1[i]; sV[i] = v[i]; }
  __syncthreads();

  const int lane = tid & 31;
  const int wave = (blockIdx.x * blockDim.x + tid) >> 5;
  const int rowBase = wave * 16;

  if (rowBase < n) {  // uniform per wave: EXEC all-ones inside
    const int m = lane & 15;
    const bool hi = lane >= 16;
    const float loMask = hi ? 0.0f : 1.0f;   // zero-pad K=3
    const int kBase = hi ? 2 * HID : 0;      // K=0 (lo) / K=2 (hi) row of W1

    // A fragment: lanes 0-15 K={0,1}, lanes 16-31 K={2,3}; row = lane%16
    const int arow = rowBase + m;
    const float a0 = input[arow * 3 + (hi ? 2 : 0)];  // single select-addr load
    const float a1 = input[arow * 3 + 1];
    v2f a;
    a.x = a0;
    a.y = loMask * a1;

    float acc[8];
#pragma unroll
    for (int r = 0; r < 8; ++r) acc[r] = 0.0f;

#pragma unroll 2
    for (int jb = 0; jb < HID; jb += 16) {
      const int col = jb + m;
      v2f b;
      b.x = sW1[kBase + col];          // ds_load, no branch
      b.y = loMask * sW1[HID + col];   // ds_load + cndmask/mul
      const float bias = sB1[col];
      const float vl = sV[col];

      v8f cfrag = {};
      cfrag = __builtin_amdgcn_wmma_f32_16x16x4_f32(
          false, a, false, b, (short)0, cfrag, false, false);

#pragma unroll
      for (int r = 0; r < 8; ++r)
        acc[r] += fast_tanhf(cfrag[r] + bias) * vl;  // v_tanh_f32 (TRANS)
    }

    // Reduce over the 16 columns (each 16-lane half shares its 8 rows).
#pragma unroll
    for (int r = 0; r < 8; ++r) {
      float s = acc[r];
      s += __shfl_xor(s, 1);
      s += __shfl_xor(s, 2);
      s += __shfl_xor(s, 4);
      s += __shfl_xor(s, 8);
      acc[r] = s;
    }
    if (m == 0) {  // lane 0 -> rows rowBase..+7, lane 16 -> rowBase+8..+15
      const float cc = *cptr;
      const int base = rowBase + (hi ? 8 : 0);
#pragma unroll
      for (int r = 0; r < 8; ++r) out[base + r] = acc[r] + cc;
    }
  }
}

// ---------------------------------------------------------------------------
extern "C" void kernel_launch(void* const* d_in, const int* in_sizes, int n_in,
                              void* d_out, int out_size, void* d_ws, size_t ws_size,
                              hipStream_t stream) {
  const float* input = (const float*)d_in[0];
  const float* eq    = (const float*)d_in[1];
  const float* W1    = (const float*)d_in[2];
  const float* b1    = (const float*)d_in[3];
  const float* W2    = (const float*)d_in[4];
  const float* b2    = (const float*)d_in[5];

  float* ws  = (float*)d_ws;
  float* S   = ws;                    // 4*64
  float* rhs = ws + 256;              // 4096 (16B-aligned for float4)
  float* v   = ws + 256 + P4;         // 1024
  float* c   = ws + 256 + P4 + HID;   // 1

  for (int i = 0; i < 4; ++i) {
    quad_S_kernel<<<1, 128, 0, stream>>>(
        eq,
        (const float*)d_in[6 + 5 * i],   // quad_x{i}
        (const float*)d_in[7 + 5 * i],   // Wq{i}a
        (const float*)d_in[8 + 5 * i],   // bq{i}a
        (const float*)d_in[9 + 5 * i],   // Wq{i}b
        (const float*)d_in[10 + 5 * i],  // bq{i}b
        S + 64 * i);
  }
  rhs_kernel<<<1, 256, 0, stream>>>(S, rhs);
  w2_rhs_kernel<<<HID + 1, 256, 0, stream>>>(W2, b2, rhs, v, c);

  const int n = in_sizes[0] / 3;  // 32768
  const int waves = (n + 15) / 16;
  const int blocks = (waves + 7) / 8;  // 8 waves (256 threads) per block
  fused_out_kernel<<<blocks, 256, 0, stream>>>(input, W1, b1, v, c,
                                               (float*)d_out, n);
}